// CustomAttn_30236569764174
// MI455X (gfx1250) — compile-verified
//
#include <hip/hip_runtime.h>
#include <math.h>

typedef __attribute__((ext_vector_type(2))) float v2f;
typedef __attribute__((ext_vector_type(4))) float v4f;
typedef __attribute__((ext_vector_type(8))) float v8f;

#define SEQ 8192
#define DIM 512  // 2*HIDDEN

// Native gfx1250 V_TANH_F32 if the toolchain exposes it; libm fallback else.
#if __has_builtin(__builtin_amdgcn_tanhf)
#define DEV_TANH(x) __builtin_amdgcn_tanhf(x)
#else
#define DEV_TANH(x) tanhf(x)
#endif

// ---------------------------------------------------------------------------
// Kernel 1: a[i] = sum_k tanh(enc[i,k]) * w[k], k < 512, via WMMA f32 16x16x4.
// One wave per 16-row tile; 8 waves per block; 64 blocks cover 512 tiles.
// A layout (16x4 f32, ISA 7.12.2): lanes 0-15 -> M=lane, VGPR0=K0, VGPR1=K1;
//                                  lanes16-31 -> M=lane-16, VGPR0=K2, VGPR1=K3.
// B layout (4x16 f32, mirrored):   lane holds column N=(lane&15), rows
//                                  K = khalf, khalf+1 (khalf = 0 or 2).
// Only column N=0 of B is non-zero (the weight vector), so D[:,0] = partial a.
// Branch-free inner loop: v2f (b64) loads, select-masked B, native tanh.
// ---------------------------------------------------------------------------
__global__ __launch_bounds__(256) void gemv_tanh_wmma(
    const float* __restrict__ enc, const float* __restrict__ w,
    float* __restrict__ a_out)
{
  const int lane  = threadIdx.x & 31;
  const int wave  = threadIdx.x >> 5;
  const int tile  = blockIdx.x * 8 + wave;
  const int r0    = tile * 16;
  const int m     = lane & 15;          // row (for A) / column (for B)
  const int khalf = (lane >> 4) * 2;    // 0 for lanes 0-15, 2 for lanes 16-31
  const bool col0 = (m == 0);

  const float* rowp = enc + (size_t)(r0 + m) * DIM + khalf;  // 8B-aligned
  const float* wp   = w + khalf;                             // 8B-aligned

  v8f c = {};
  for (int k0 = 0; k0 < DIM; k0 += 16) {
#pragma unroll
    for (int u = 0; u < 4; ++u) {
      const int k = k0 + 4 * u;
      const v2f av = *(const v2f*)(rowp + k);   // global_load_b64
      const v2f wv = *(const v2f*)(wp + k);     // global_load_b64 (uniform-ish)
      v2f A;
      A.x = DEV_TANH(av.x);
      A.y = DEV_TANH(av.y);
      v2f B;
      B.x = col0 ? wv.x : 0.0f;   // v_cndmask, no EXEC divergence
      B.y = col0 ? wv.y : 0.0f;
      c = __builtin_amdgcn_wmma_f32_16x16x4_f32(
          /*neg_a=*/false, A, /*neg_b=*/false, B,
          /*c_mod=*/(short)0, c, /*reuse_a=*/false, /*reuse_b=*/false);
    }
  }

  // C/D layout: VGPR v, lanes 0-15 -> (M=v, N=lane); lanes 16-31 -> (M=v+8, N=lane-16).
  // Column N=0 lives in lane 0 (M=0..7) and lane 16 (M=8..15).
  if (m == 0) {
    const int mbase = (lane >> 4) * 8;
#pragma unroll
    for (int v = 0; v < 8; ++v) a_out[r0 + mbase + v] = c[v];
  }
}

// ---------------------------------------------------------------------------
// Kernel 2: p = softmax(a) over all 8192 entries. Single 1024-thread block,
// 8 elements per thread. (b[j] + bias cancel exactly in the axis-0 softmax.)
// ---------------------------------------------------------------------------
__global__ __launch_bounds__(1024) void softmax_vec(
    const float* __restrict__ a, float* __restrict__ p)
{
  __shared__ float red[32];
  const int tid = threadIdx.x;

  float local[8];
  float mx = -INFINITY;
#pragma unroll
  for (int i = 0; i < 8; ++i) {
    local[i] = a[tid * 8 + i];
    mx = fmaxf(mx, local[i]);
  }
  // wave32 max reduction
#pragma unroll
  for (int off = 16; off > 0; off >>= 1)
    mx = fmaxf(mx, __shfl_xor(mx, off, 32));
  if ((tid & 31) == 0) red[tid >> 5] = mx;
  __syncthreads();
  {
    float v = red[tid & 31];  // each wave reduces all 32 partials
#pragma unroll
    for (int off = 16; off > 0; off >>= 1)
      v = fmaxf(v, __shfl_xor(v, off, 32));
    mx = v;
  }
  __syncthreads();  // red[] about to be reused

  float sum = 0.0f;
#pragma unroll
  for (int i = 0; i < 8; ++i) {
    local[i] = __expf(local[i] - mx);
    sum += local[i];
  }
#pragma unroll
  for (int off = 16; off > 0; off >>= 1)
    sum += __shfl_xor(sum, off, 32);
  if ((tid & 31) == 0) red[tid >> 5] = sum;
  __syncthreads();
  {
    float v = red[tid & 31];
#pragma unroll
    for (int off = 16; off > 0; off >>= 1)
      v += __shfl_xor(v, off, 32);
    sum = v;
  }

  const float inv = 1.0f / sum;
#pragma unroll
  for (int i = 0; i < 8; ++i) p[tid * 8 + i] = local[i] * inv;
}

// ---------------------------------------------------------------------------
// Kernel 3: out[i, j] = p[i] for all j. 256 MB streaming write -> the only
// cost that matters (~11 us at 23.3 TB/s). 128-bit non-temporal stores.
// One block per row; 256 threads x 8 iterations x 16 B = 32 KB row.
// ---------------------------------------------------------------------------
__global__ __launch_bounds__(256) void bcast_rows(
    const float* __restrict__ p, v4f* __restrict__ out)
{
  const int row  = blockIdx.x;
  const float v  = p[row];
  const v4f   q  = {v, v, v, v};
  v4f* rowp = out + (size_t)row * (SEQ / 4);
#pragma unroll
  for (int j = threadIdx.x; j < SEQ / 4; j += 256)
    __builtin_nontemporal_store(q, rowp + j);
}

// ---------------------------------------------------------------------------
extern "C" void kernel_launch(void* const* d_in, const int* in_sizes, int n_in,
                              void* d_out, int out_size, void* d_ws, size_t ws_size,
                              hipStream_t stream) {
  (void)in_sizes; (void)n_in; (void)out_size; (void)ws_size;
  const float* enc  = (const float*)d_in[0];  // [8192, 512] f32
  const float* w    = (const float*)d_in[1];  // [1024] f32
  // d_in[2] = bias: cancels exactly in the axis-0 softmax; unused.

  float* a = (float*)d_ws;        // [8192]
  float* p = a + SEQ;             // [8192]

  gemv_tanh_wmma<<<SEQ / 16 / 8, 256, 0, stream>>>(enc, w, a);
  softmax_vec<<<1, 1024, 0, stream>>>(a, p);
  bcast_rows<<<SEQ, 256, 0, stream>>>(p, (v4f*)d_out);
}